// OhemBCELoss_53584011985662
// MI455X (gfx1250) — compile-verified
//
#include <hip/hip_runtime.h>

#define IGNORE_LABEL (-100.0f)
#define LOG_CLAMP    (-100.0f)
#define OHEM_THRESH  (0.7f)
#define OHEM_MIN_KEPT 100000LL

#if defined(__AMDGCN__) && __has_builtin(__builtin_amdgcn_global_load_async_to_lds_b128) && __has_builtin(__builtin_amdgcn_s_wait_asynccnt)
#define OHEM_ASYNC 1
#else
#define OHEM_ASYNC 0
#endif

// ---- monotone float <-> uint key (order-preserving for all floats) ----
__device__ __forceinline__ unsigned keyOf(float f) {
  unsigned u = __float_as_uint(f);
  return (u & 0x80000000u) ? ~u : (u | 0x80000000u);
}
__device__ __forceinline__ float unKey(unsigned k) {
  return __uint_as_float((k & 0x80000000u) ? (k & 0x7FFFFFFFu) : ~k);
}

// =====================================================================
// Pass 1: histogram of high 16 key bits, per-WG 256KB LDS histogram
// =====================================================================
__global__ __launch_bounds__(1024) void ohem_hist_hi(
    const float* __restrict__ p, const float* __restrict__ t,
    unsigned* __restrict__ ghist, int n4, int n) {
  __shared__ unsigned lhist[65536];          // 256 KB — CDNA5 WGP has 320 KB
  for (int i = threadIdx.x; i < 65536; i += 1024) lhist[i] = 0u;
  __syncthreads();

  const float4* p4 = (const float4*)p;
  const float4* t4 = (const float4*)t;
  int gid = blockIdx.x * 1024 + threadIdx.x;
  int stride = gridDim.x * 1024;
  for (int i = gid; i < n4; i += stride) {
    float4 pv = p4[i];
    float4 tv = t4[i];
    if (tv.x != IGNORE_LABEL) atomicAdd(&lhist[keyOf(pv.x) >> 16], 1u);
    if (tv.y != IGNORE_LABEL) atomicAdd(&lhist[keyOf(pv.y) >> 16], 1u);
    if (tv.z != IGNORE_LABEL) atomicAdd(&lhist[keyOf(pv.z) >> 16], 1u);
    if (tv.w != IGNORE_LABEL) atomicAdd(&lhist[keyOf(pv.w) >> 16], 1u);
  }
  // scalar tail
  for (int i = (n4 << 2) + gid; i < n; i += stride) {
    if (t[i] != IGNORE_LABEL) atomicAdd(&lhist[keyOf(p[i]) >> 16], 1u);
  }
  __syncthreads();
  // flush: skip empty bins (uniform(0,1) data concentrates in ~1.3K bins)
  for (int i = threadIdx.x; i < 65536; i += 1024) {
    unsigned c = lhist[i];
    if (c) atomicAdd(&ghist[i], c);
  }
}

// =====================================================================
// Scan a 65536-bin histogram to locate the bin containing rank k.
// level 0: k = min(MIN_KEPT, n_valid-1); writes sel[0]=bin, sel[1]=rank-in-bin,
//          sel[2]=n_valid.
// level 1: k = sel[1]; writes final threshold (float bits) to sel[3].
// =====================================================================
__global__ __launch_bounds__(1024) void ohem_scan(
    const unsigned* __restrict__ hist, unsigned* __restrict__ sel, int level) {
  __shared__ unsigned long long part[1024];
  int tid = threadIdx.x;
  unsigned long long s = 0;
  int base = tid * 64;
  for (int j = 0; j < 64; ++j) s += hist[base + j];
  part[tid] = s;
  __syncthreads();
  if (tid == 0) {
    unsigned long long total = 0;
    for (int i = 0; i < 1024; ++i) total += part[i];
    unsigned long long kk;
    if (level == 0) {
      long long k = (long long)total - 1;
      if (k > OHEM_MIN_KEPT) k = OHEM_MIN_KEPT;
      if (k < 0) k = 0;
      kk = (unsigned long long)k;
      sel[2] = (unsigned)total;   // n_valid
    } else {
      kk = (unsigned long long)sel[1];
    }
    unsigned long long c = 0;
    int bin = 65535;
    for (int i = 0; i < 1024; ++i) {
      if (c + part[i] > kk) {
        unsigned long long cc = c;
        for (int j = 0; j < 64; ++j) {
          unsigned h = hist[i * 64 + j];
          if (cc + h > kk) { bin = i * 64 + j; c = cc; goto done; }
          cc += h;
        }
      }
      c += part[i];
    }
  done:
    if (level == 0) {
      sel[0] = (unsigned)bin;
      sel[1] = (unsigned)(kk - c);      // rank within chosen bin
    } else {
      unsigned key = (sel[0] << 16) | (unsigned)bin;
      float thr = fmaxf(unKey(key), OHEM_THRESH);
      ((float*)sel)[3] = thr;
    }
  }
}

// =====================================================================
// Pass 2: histogram of low 16 key bits for elements in the selected high
// bin. Only ~n/65536 elements hit -> direct global atomics are cheap.
// =====================================================================
__global__ __launch_bounds__(256) void ohem_hist_lo(
    const float* __restrict__ p, const float* __restrict__ t,
    const unsigned* __restrict__ sel, unsigned* __restrict__ ghist2,
    int n4, int n) {
  unsigned bin1 = sel[0];
  const float4* p4 = (const float4*)p;
  const float4* t4 = (const float4*)t;
  int gid = blockIdx.x * 256 + threadIdx.x;
  int stride = gridDim.x * 256;
  for (int i = gid; i < n4; i += stride) {
    float4 pv = p4[i];
    float4 tv = t4[i];
    unsigned k;
    if (tv.x != IGNORE_LABEL) { k = keyOf(pv.x); if ((k >> 16) == bin1) atomicAdd(&ghist2[k & 0xFFFFu], 1u); }
    if (tv.y != IGNORE_LABEL) { k = keyOf(pv.y); if ((k >> 16) == bin1) atomicAdd(&ghist2[k & 0xFFFFu], 1u); }
    if (tv.z != IGNORE_LABEL) { k = keyOf(pv.z); if ((k >> 16) == bin1) atomicAdd(&ghist2[k & 0xFFFFu], 1u); }
    if (tv.w != IGNORE_LABEL) { k = keyOf(pv.w); if ((k >> 16) == bin1) atomicAdd(&ghist2[k & 0xFFFFu], 1u); }
  }
  for (int i = (n4 << 2) + gid; i < n; i += stride) {
    if (t[i] != IGNORE_LABEL) {
      unsigned k = keyOf(p[i]);
      if ((k >> 16) == bin1) atomicAdd(&ghist2[k & 0xFFFFu], 1u);
    }
  }
}

// =====================================================================
// Pass 3: masked weighted BCE partial sums.
// CDNA5 path: double-buffered GLOBAL_LOAD_ASYNC_TO_LDS_B128 staging of
// p/t/w tiles, synchronized with s_wait_asynccnt. Each lane stages and
// reads only its own 16B per array -> no barrier needed, only the async
// counter wait (async loads complete in order).
// Deterministic: fixed grid, fixed per-thread order, tree reduction.
// =====================================================================
__device__ __forceinline__ void ohem_acc(float pv, float tv, float wv, float thr,
                                         double& ls, unsigned long long& lc) {
  if (tv != IGNORE_LABEL && pv < thr) {
    float lp  = fmaxf(logf(pv),    LOG_CLAMP);
    float l1  = fmaxf(log1pf(-pv), LOG_CLAMP);
    float loss = -wv * (tv * lp + (1.0f - tv) * l1);
    ls += (double)loss;
    lc += 1ull;
  }
}

#if OHEM_ASYNC
typedef int ohem_v4i __attribute__((ext_vector_type(4)));

__device__ __forceinline__ void ohem_cp16(const float4* g, float4* l) {
  __builtin_amdgcn_global_load_async_to_lds_b128(
      (__attribute__((address_space(1))) ohem_v4i*)(void*)g,
      (__attribute__((address_space(3))) ohem_v4i*)(void*)l,
      0, 0);
}
#endif

#define OHEM_TILE 256

__global__ __launch_bounds__(256) void ohem_loss(
    const float* __restrict__ p, const float* __restrict__ t,
    const float* __restrict__ w, const unsigned* __restrict__ sel,
    double* __restrict__ psum, unsigned long long* __restrict__ pcnt,
    int n4, int n) {
  float thr = __uint_as_float(sel[3]);
  const float4* p4 = (const float4*)p;
  const float4* t4 = (const float4*)t;
  const float4* w4 = (const float4*)w;
  int tid = threadIdx.x;
  double ls = 0.0;
  unsigned long long lc = 0ull;

#if OHEM_ASYNC
  __shared__ float4 sp[2][OHEM_TILE];
  __shared__ float4 stg[2][OHEM_TILE];
  __shared__ float4 swt[2][OHEM_TILE];

  int nTiles = n4 / OHEM_TILE;          // full tiles of 256 float4s
  int tile0 = blockIdx.x;
  int tstride = gridDim.x;

  if (tile0 < nTiles) {
    // prologue: stage first tile into buffer 0
    {
      int base = tile0 * OHEM_TILE + tid;
      ohem_cp16(p4 + base, &sp[0][tid]);
      ohem_cp16(t4 + base, &stg[0][tid]);
      ohem_cp16(w4 + base, &swt[0][tid]);
    }
    int s = 0;
    for (int it = tile0; it < nTiles; it += tstride) {
      int nx = it + tstride;
      if (nx < nTiles) {
        int base = nx * OHEM_TILE + tid;
        ohem_cp16(p4 + base, &sp[s ^ 1][tid]);
        ohem_cp16(t4 + base, &stg[s ^ 1][tid]);
        ohem_cp16(w4 + base, &swt[s ^ 1][tid]);
        __builtin_amdgcn_s_wait_asynccnt(3);   // current stage (3 ops) done
      } else {
        __builtin_amdgcn_s_wait_asynccnt(0);   // drain
      }
      float4 pv = sp[s][tid];
      float4 tv = stg[s][tid];
      float4 wv = swt[s][tid];
      ohem_acc(pv.x, tv.x, wv.x, thr, ls, lc);
      ohem_acc(pv.y, tv.y, wv.y, thr, ls, lc);
      ohem_acc(pv.z, tv.z, wv.z, thr, ls, lc);
      ohem_acc(pv.w, tv.w, wv.w, thr, ls, lc);
      s ^= 1;
    }
  }
  // leftover float4s beyond the full tiles
  {
    int gid = blockIdx.x * 256 + tid;
    int stride = gridDim.x * 256;
    for (int i = nTiles * OHEM_TILE + gid; i < n4; i += stride) {
      float4 pv = p4[i];
      float4 tv = t4[i];
      float4 wv = w4[i];
      ohem_acc(pv.x, tv.x, wv.x, thr, ls, lc);
      ohem_acc(pv.y, tv.y, wv.y, thr, ls, lc);
      ohem_acc(pv.z, tv.z, wv.z, thr, ls, lc);
      ohem_acc(pv.w, tv.w, wv.w, thr, ls, lc);
    }
  }
#else
  {
    int gid = blockIdx.x * 256 + tid;
    int stride = gridDim.x * 256;
    for (int i = gid; i < n4; i += stride) {
      if (i + stride < n4) {
        __builtin_prefetch(&p4[i + stride], 0, 0);
        __builtin_prefetch(&t4[i + stride], 0, 0);
        __builtin_prefetch(&w4[i + stride], 0, 0);
      }
      float4 pv = p4[i];
      float4 tv = t4[i];
      float4 wv = w4[i];
      ohem_acc(pv.x, tv.x, wv.x, thr, ls, lc);
      ohem_acc(pv.y, tv.y, wv.y, thr, ls, lc);
      ohem_acc(pv.z, tv.z, wv.z, thr, ls, lc);
      ohem_acc(pv.w, tv.w, wv.w, thr, ls, lc);
    }
  }
#endif
  // scalar tail
  {
    int gid = blockIdx.x * 256 + tid;
    int stride = gridDim.x * 256;
    for (int i = (n4 << 2) + gid; i < n; i += stride) {
      ohem_acc(p[i], t[i], w[i], thr, ls, lc);
    }
  }

  __shared__ double ssum[256];
  __shared__ unsigned long long scnt[256];
  ssum[tid] = ls;
  scnt[tid] = lc;
  __syncthreads();
  for (int off = 128; off > 0; off >>= 1) {
    if (tid < off) {
      ssum[tid] += ssum[tid + off];
      scnt[tid] += scnt[tid + off];
    }
    __syncthreads();
  }
  if (tid == 0) {
    psum[blockIdx.x] = ssum[0];
    pcnt[blockIdx.x] = scnt[0];
  }
}

__global__ __launch_bounds__(256) void ohem_final(
    const double* __restrict__ psum, const unsigned long long* __restrict__ pcnt,
    float* __restrict__ out, int nblk) {
  __shared__ double ssum[256];
  __shared__ unsigned long long scnt[256];
  int tid = threadIdx.x;
  double s = 0.0;
  unsigned long long c = 0ull;
  for (int i = tid; i < nblk; i += 256) { s += psum[i]; c += pcnt[i]; }
  ssum[tid] = s;
  scnt[tid] = c;
  __syncthreads();
  for (int off = 128; off > 0; off >>= 1) {
    if (tid < off) {
      ssum[tid] += ssum[tid + off];
      scnt[tid] += scnt[tid + off];
    }
    __syncthreads();
  }
  if (tid == 0) out[0] = (float)(ssum[0] / (double)scnt[0]);
}

// =====================================================================
// Workspace layout (bytes):
//   [0      , 262144)  hist1 (65536 x u32)
//   [262144 , 524288)  hist2 (65536 x u32)
//   [524288 , 524304)  sel   (4 x u32: bin1, rank, n_valid, thr_bits)
//   [524352 , 540736)  psum  (2048 x double)
//   [540736 , 557120)  pcnt  (2048 x u64)
// =====================================================================
extern "C" void kernel_launch(void* const* d_in, const int* in_sizes, int n_in,
                              void* d_out, int out_size, void* d_ws, size_t ws_size,
                              hipStream_t stream) {
  const float* p = (const float*)d_in[0];
  const float* t = (const float*)d_in[1];
  const float* w = (const float*)d_in[2];
  int n  = in_sizes[0];
  int n4 = n >> 2;

  char* base = (char*)d_ws;
  unsigned* hist1 = (unsigned*)base;
  unsigned* hist2 = (unsigned*)(base + 262144);
  unsigned* sel   = (unsigned*)(base + 524288);
  double* psum    = (double*)(base + 524352);
  unsigned long long* pcnt = (unsigned long long*)(base + 540736);

  const int LOSS_BLOCKS = 2048;

  (void)hipMemsetAsync(d_ws, 0, 524352, stream);  // hist1 + hist2 + sel

  ohem_hist_hi<<<512, 1024, 0, stream>>>(p, t, hist1, n4, n);
  ohem_scan<<<1, 1024, 0, stream>>>(hist1, sel, 0);
  ohem_hist_lo<<<2048, 256, 0, stream>>>(p, t, sel, hist2, n4, n);
  ohem_scan<<<1, 1024, 0, stream>>>(hist2, sel, 1);
  ohem_loss<<<LOSS_BLOCKS, 256, 0, stream>>>(p, t, w, sel, psum, pcnt, n4, n);
  ohem_final<<<1, 256, 0, stream>>>(psum, pcnt, (float*)d_out, LOSS_BLOCKS);
}